// RadialTokenizer_64682207478582
// MI455X (gfx1250) — compile-verified
//
#include <hip/hip_runtime.h>
#include <hip/hip_bf16.h>
#include <math.h>

// ---------------------------------------------------------------------------
// RadialTokenizer for MI455X (gfx1250).
// HBM-bound: 805 MB in / 0.6 MB out -> ~35us floor @ 23.3 TB/s.
// Path: GLOBAL_LOAD_ASYNC_TO_LDS_B128 double-buffered per wave (ASYNCcnt),
// 256-bin LDS histogram (values are exact ints 0..255 after floor(x*255)),
// mean/std/median all derived exactly from the histogram.
// v2: float4 LDS reads (4x fewer ds_load instrs on the shared DS pipe),
//     wave-parallel histogram reduction (shfl scan) to kill the serial tail.
// ---------------------------------------------------------------------------

// 4 async B128 loads = one 2KB chunk per wave (32 lanes x 16B x 4).
// Per CDNA5 ISA, IOFFSET is added to BOTH the global address and
// the per-lane LDS address, so one (lds,vaddr) pair covers all 4 instructions.
#define ASYNC_LOAD4(ldsoff, gptr)                                              \
  asm volatile("global_load_async_to_lds_b128 %0, %1, off\n\t"                 \
               "global_load_async_to_lds_b128 %0, %1, off offset:512\n\t"      \
               "global_load_async_to_lds_b128 %0, %1, off offset:1024\n\t"     \
               "global_load_async_to_lds_b128 %0, %1, off offset:1536"         \
               :: "v"(ldsoff), "v"(gptr) : "memory")

#define WAIT_ASYNCCNT_4() asm volatile("s_wait_asynccnt 0x4" ::: "memory")
#define WAIT_ASYNCCNT_0() asm volatile("s_wait_asynccnt 0x0" ::: "memory")

__device__ __forceinline__ unsigned lds_byte_off(const void* p) {
  // Generic pointers to LDS carry the byte offset in the low 32 bits.
  return (unsigned)(uintptr_t)p;
}

// 512-pixel chunk: lane reads 4 float4s (ds_load_b128, conflict-free:
// 32 lanes x 16B consecutive). 4 consecutive pixels share a row (128%4==0),
// so dy^2 is computed once per quad.
__device__ __forceinline__ void process_chunk(const float* __restrict__ buf,
                                              int pix0, int l,
                                              unsigned* __restrict__ hist) {
  const float4* __restrict__ buf4 = (const float4*)buf;
#pragma unroll
  for (int j = 0; j < 4; ++j) {
    const int q = j * 32 + l;            // float4 index in chunk, 0..127
    const float4 v4 = buf4[q];
    float vs[4] = {v4.x, v4.y, v4.z, v4.w};
    const int p0 = pix0 + q * 4;         // first pixel of the quad
    const int dy = (p0 >> 7) - 64;
    const int dy2 = dy * dy;
    const int dx0 = (p0 & 127) - 64;
#pragma unroll
    for (int e = 0; e < 4; ++e) {
      const int dx = dx0 + e;
      const int d2 = dy2 + dx * dx;
      if (d2 > 0 && d2 <= 4096) {        // rings (0,16],(16,32],(32,48],(48,64]
        const int ring = (d2 > 256) + (d2 > 1024) + (d2 > 2304);
        int bin = (int)(vs[e] * 255.0f); // v in [0,1): trunc == floor
        bin = bin > 255 ? 255 : bin;
        atomicAdd(&hist[(ring << 8) + bin], 1u);   // ds_add_u32
      }
    }
  }
}

__global__ __launch_bounds__(256) void radial_tokenizer_kernel(
    const float* __restrict__ in, float* __restrict__ out) {
  // 8 waves x 2 buffers x 512 floats = 32KB staging, +4KB histogram.
  __shared__ __align__(16) float stage[8192];
  __shared__ unsigned hist[1024];

  const int t = threadIdx.x;
  const int w = t >> 5;                  // wave id 0..7 (wave32)
  const int l = t & 31;                  // lane id
  const int bc = blockIdx.x;             // (b*3 + c)
  const int b = bc / 3;
  const int c = bc - b * 3;

  // Wave w streams pixels [w*2048, w*2048+2048) of this (b,c) image.
  const char* gbase =
      (const char*)(in + (size_t)bc * 16384) + (size_t)w * 8192 + (size_t)l * 16;
  float* const wbuf = &stage[w * 1024];
  const unsigned lds0 = lds_byte_off(&stage[w * 1024 + l * 4]); // buffer 0
  const unsigned lds1 = lds0 + 2048;                            // buffer 1

  // Prime the pipeline: chunk0 -> buf0, chunk1 -> buf1.
  ASYNC_LOAD4(lds0, gbase);
  ASYNC_LOAD4(lds1, gbase + 2048);

  // Zero the histogram while the first loads are in flight.
#pragma unroll
  for (int i = 0; i < 4; ++i) hist[t * 4 + i] = 0u;
  __syncthreads();

  const int pbase = w * 2048;

  // k=0: need chunk0 done; {chunk0,chunk1} outstanding -> wait <=4.
  WAIT_ASYNCCNT_4();
  process_chunk(wbuf, pbase, l, hist);
  ASYNC_LOAD4(lds0, gbase + 4096);       // chunk2 -> buf0
  // k=1
  WAIT_ASYNCCNT_4();
  process_chunk(wbuf + 512, pbase + 512, l, hist);
  ASYNC_LOAD4(lds1, gbase + 6144);       // chunk3 -> buf1
  // k=2
  WAIT_ASYNCCNT_4();
  process_chunk(wbuf, pbase + 1024, l, hist);
  // k=3: only chunk3 outstanding -> wait <=0.
  WAIT_ASYNCCNT_0();
  process_chunk(wbuf + 512, pbase + 1536, l, hist);

  __syncthreads();

  // Wave w (w<4) reduces ring w's 256-bin histogram in parallel:
  // lane owns 8 bins; shfl butterflies for s1/s2, n from the scan total; shfl_up inclusive scan
  // of per-lane counts localizes the two order statistics for the median.
  if (w < 4) {
    const unsigned* __restrict__ h = &hist[w << 8];
    unsigned hb[8];
    unsigned cnt = 0, ls1 = 0;
    unsigned long long ls2 = 0;
#pragma unroll
    for (int e = 0; e < 8; ++e) {
      const int v = l * 8 + e;
      hb[e] = h[v];
      cnt += hb[e];
      ls1 += (unsigned)v * hb[e];
      ls2 += (unsigned long long)(v * v) * hb[e];
    }
    // Inclusive scan of cnt across the wave.
    unsigned inc = cnt;
#pragma unroll
    for (int d = 1; d < 32; d <<= 1) {
      const unsigned up = __shfl_up(inc, d, 32);
      if (l >= d) inc += up;
    }
    const unsigned n = __shfl(inc, 31, 32);
    // Butterfly reductions for s1 (u32) and s2 (u64).
    unsigned s1 = ls1;
    unsigned long long s2 = ls2;
#pragma unroll
    for (int d = 16; d >= 1; d >>= 1) {
      s1 += __shfl_xor(s1, d, 32);
      s2 += __shfl_xor(s2, d, 32);
    }
    // Order statistics k1=(n-1)/2, k2=n/2 (0-indexed).
    const unsigned k1 = (n - 1u) >> 1, k2 = n >> 1;
    const unsigned excl = inc - cnt;
    int m1loc = -1, m2loc = -1;
    if (excl <= k1 && k1 < inc) {
      unsigned cum = excl;
#pragma unroll
      for (int e = 0; e < 8; ++e) {
        cum += hb[e];
        if (m1loc < 0 && cum > k1) m1loc = l * 8 + e;
      }
    }
    if (excl <= k2 && k2 < inc) {
      unsigned cum = excl;
#pragma unroll
      for (int e = 0; e < 8; ++e) {
        cum += hb[e];
        if (m2loc < 0 && cum > k2) m2loc = l * 8 + e;
      }
    }
    int m1 = m1loc, m2 = m2loc;
#pragma unroll
    for (int d = 16; d >= 1; d >>= 1) {
      m1 = max(m1, __shfl_xor(m1, d, 32));
      m2 = max(m2, __shfl_xor(m2, d, 32));
    }
    if (l == 0) {
      const double dn = (double)n;
      const double mean = (double)s1 / dn;
      double var = ((double)s2 - (double)s1 * (double)s1 / dn) / dn;
      if (var < 0.0) var = 0.0;
      // out layout (B, 4, 9): [mean c0..c2 | std c0..c2 | med c0..c2]
      float* o = out + ((size_t)b * 4 + (size_t)w) * 9;
      o[c] = (float)mean;
      o[3 + c] = (float)sqrt(var);
      o[6 + c] = 0.5f * (float)(m1 + m2);
    }
  }
}

extern "C" void kernel_launch(void* const* d_in, const int* in_sizes, int n_in,
                              void* d_out, int out_size, void* d_ws, size_t ws_size,
                              hipStream_t stream) {
  (void)n_in; (void)out_size; (void)d_ws; (void)ws_size;
  const float* in = (const float*)d_in[0];
  float* out = (float*)d_out;
  const int B = in_sizes[0] / (3 * 128 * 128);   // 4096
  radial_tokenizer_kernel<<<B * 3, 256, 0, stream>>>(in, out);
}